// LayeredSNN_47656957116818
// MI455X (gfx1250) — compile-verified
//
#include <hip/hip_runtime.h>

// ---------------------------------------------------------------------------
// LayeredSNN forward on gfx1250 (MI455X).
// Dominant cost: 3x GEMV  pot[o] = sum_k W[o,k]*trace[k],  W: [2048,16384] f32.
// Memory bound: 403 MB of fp32 weights @ 23.3 TB/s ~= 17 us floor (AI ~ 0.5
// FLOP/byte), so the design goal is pure streaming. GEMV is mapped onto
// V_WMMA_F32_16X16X4_F32 with the trace broadcast across all 16 B columns
// (every D column equals the desired dot product). The 64 KB trace vector is
// staged into LDS by the Tensor Data Mover (one tensor_load_to_lds per block,
// TENSORcnt-tracked) and read back as ds_load_b64 B fragments.
// Feedback potentials are identically zero (fresh zero spike histories).
// ---------------------------------------------------------------------------

typedef __attribute__((ext_vector_type(2))) float v2f;
typedef __attribute__((ext_vector_type(8))) float v8f;
typedef __attribute__((ext_vector_type(4))) unsigned int v4u;
typedef __attribute__((ext_vector_type(8))) int v8i;
typedef __attribute__((ext_vector_type(4))) int v4i;

#define NN      2048
#define NB      8
#define KTOT    (NN * NB)          // 16384
#define KSPLIT  8
#define KCHUNK  (KTOT / KSPLIT)    // 2048
#define PI_F    3.14159265358979323846f

// ---- Layer-0 trace: trace[i,b] = sum_t inp[i, 9-t] * FF_FILT[t,b] ----------
__global__ __launch_bounds__(256) void snn_trace0(const float* __restrict__ inp,
                                                  float* __restrict__ trace) {
  int idx = blockIdx.x * 256 + threadIdx.x;       // 16384 work items
  if (idx >= KTOT) return;
  int i = idx >> 3;
  int b = idx & 7;
  float acc = 0.0f;
#pragma unroll
  for (int t = 0; t < 10; ++t) {
    // Pillow raised-cosine basis, MU = 0.5 (fixed constant of the module)
    float arg = 0.5f * PI_F * log1pf((float)t) - 0.5f * PI_F * (float)b;
    arg = fminf(fmaxf(arg, -PI_F), PI_F);
    float f = 0.5f * (1.0f + __cosf(arg));
    acc += inp[i * 10 + (9 - t)] * f;             // reversed time axis
  }
  trace[idx] = acc;
}

// ---- TDM: DMA the 64 KB trace vector (1-D, 16384 x f32) into LDS offset 0 --
__device__ __forceinline__ void tdm_load_trace_to_lds(const float* src,
                                                      unsigned lds_off) {
  unsigned long long ga = (unsigned long long)(size_t)src;
  // D# group 0: count=1 (valid user descriptor), lds_addr, global_addr, type=2
  v4u g0;
  g0[0] = 1u;                                      // [1:0] count = 1
  g0[1] = lds_off;                                 // [63:32] lds_addr
  g0[2] = (unsigned)(ga & 0xFFFFFFFFu);            // [95:64] global_addr lo
  g0[3] = (unsigned)((ga >> 32) & 0x01FFFFFFu)     // [120:96] global_addr hi
        | (2u << 30);                              // [127:126] type = 2
  // D# group 1: data_size=4B, tensor_dim0 = tile_dim0 = 16384, 1 row
  v8i g1;
  g1[0] = 0x00020000;                 // workgroup_mask=0, data_size=2 (4B)
  g1[1] = (int)(0x4000u << 16);       // tensor_dim0[15:0]=16384 @ bits 63:48
  g1[2] = (int)(1u << 16);            // tensor_dim0 hi=0; tensor_dim1=1
  g1[3] = (int)(0x4000u << 16);       // tensor_dim1 hi=0; tile_dim0=16384
  g1[4] = 1;                          // tile_dim1=1, tile_dim2=0 (unused)
  g1[5] = 16384;                      // tensor_dim0_stride lo
  g1[6] = 0;                          // stride hi / dim1_stride lo
  g1[7] = 0;                          // dim1_stride hi
  v4i z4 = {0, 0, 0, 0};
#if __clang_major__ >= 23
  v8i z8 = {0, 0, 0, 0, 0, 0, 0, 0};
  __builtin_amdgcn_tensor_load_to_lds(g0, g1, z4, z4, z8, 0);
#else
  __builtin_amdgcn_tensor_load_to_lds(g0, g1, z4, z4, 0);
#endif
}

// ---- WMMA split-K GEMV: partial[wave][row] = W[row, chunk] . trace[chunk] --
__global__ __launch_bounds__(256) void snn_gemv_wmma(const float* __restrict__ W,
                                                     const float* __restrict__ trace,
                                                     float* __restrict__ partial) {
  __shared__ float tr[KTOT];                      // 64 KB of 320 KB WGP LDS

  const int tid = threadIdx.x;
  if (tid < 32) {                                 // wave 0 issues one TDM op
    tdm_load_trace_to_lds(trace, (unsigned)(size_t)tr);
    __builtin_amdgcn_s_wait_tensorcnt(0);
  }
  __syncthreads();                                // publish LDS to all 8 waves

  const int wave  = tid >> 5;                     // 0..7  -> K chunk
  const int lane  = tid & 31;
  const int tile  = blockIdx.x;                   // 0..127 -> 16-row tile
  const int m     = lane & 15;                    // A row within tile
  const int khalf = lane >> 4;                    // ISA A layout: lanes>=16 hold K+2/K+3

  const float* __restrict__ wrow =
      W + (size_t)(tile * 16 + m) * KTOT + wave * KCHUNK + 2 * khalf;
  const float* tp = tr + wave * KCHUNK + 2 * khalf;

  v8f c = {0.f, 0.f, 0.f, 0.f, 0.f, 0.f, 0.f, 0.f};

  for (int k0 = 0; k0 < KCHUNK; k0 += 64) {
    __builtin_prefetch(wrow + k0 + 64, 0, 1);     // global_prefetch next strip
#pragma unroll
    for (int k = k0; k < k0 + 64; k += 4) {
      v2f a, b;
      a.x = wrow[k];                              // K + 2*khalf + 0
      a.y = wrow[k + 1];                          // K + 2*khalf + 1
      // Broadcast trace across all 16 N columns: B[k][n] = trace[k] for all n.
      b.x = tp[k];
      b.y = tp[k + 1];
      c = __builtin_amdgcn_wmma_f32_16x16x4_f32(false, a, false, b,
                                                (short)0, c, false, false);
    }
  }

  // D layout (16x16 f32): lanes 0-15 VGPR r -> D[r][lane]; lanes 16-31 -> D[8+r][lane-16].
  // All columns are equal, so lanes 0..7 deliver rows 0..7, lanes 16..23 rows 8..15.
  const int rsel = lane & 7;
  float val = 0.0f;
#pragma unroll
  for (int r = 0; r < 8; ++r) val = (rsel == r) ? c[r] : val;
  const bool active = (lane < 8) || (lane >= 16 && lane < 24);
  if (active) {
    int row = tile * 16 + ((lane >> 4) << 3) + rsel;
    partial[wave * NN + row] = val;               // per-wave slot: no atomics
  }
}

// ---- Reduce partials, sigmoid, Bernoulli draw, emit next-layer trace -------
__global__ __launch_bounds__(256) void snn_act(const float* __restrict__ partial,
                                               const float* __restrict__ bias,
                                               float* __restrict__ p_out,
                                               float* __restrict__ s_out,
                                               float* __restrict__ trace_next,
                                               unsigned seed) {
  int i = blockIdx.x * 256 + threadIdx.x;
  if (i >= NN) return;
  float pot = bias[i];                            // fb_pot == 0 (zero histories)
#pragma unroll
  for (int r = 0; r < KSPLIT; ++r) pot += partial[r * NN + i];
  float p = 1.0f / (1.0f + __expf(-pot));

  // Deterministic splitmix64 hash -> uniform [0,1)
  unsigned long long x = (((unsigned long long)seed) << 32) | (unsigned)i;
  x += 0x9E3779B97F4A7C15ull;
  x ^= x >> 30; x *= 0xBF58476D1CE4E5B9ull;
  x ^= x >> 27; x *= 0x94D049BB133111EBull;
  x ^= x >> 31;
  float u = (float)(unsigned)(x >> 40) * (1.0f / 16777216.0f);
  float s = (u < p) ? 1.0f : 0.0f;

  p_out[i] = p;
  s_out[i] = s;

  if (trace_next != nullptr) {
    // inp_hist = [0, 0, s] reversed -> trace = s * FF_FILT[0,:] = s*[1, .5, 0...]
    float* t = trace_next + i * NB;
    t[0] = s;
    t[1] = 0.5f * s;
#pragma unroll
    for (int b = 2; b < NB; ++b) t[b] = 0.0f;
  }
}

extern "C" void kernel_launch(void* const* d_in, const int* in_sizes, int n_in,
                              void* d_out, int out_size, void* d_ws, size_t ws_size,
                              hipStream_t stream) {
  const float* inp   = (const float*)d_in[0];   // [2048,10]
  const float* ff_w0 = (const float*)d_in[1];   // [2048,2048,8]
  const float* b0    = (const float*)d_in[3];
  const float* ff_w1 = (const float*)d_in[4];
  const float* b1    = (const float*)d_in[6];
  const float* ff_w2 = (const float*)d_in[7];
  const float* b2    = (const float*)d_in[9];
  // fb_w0/1/2 (d_in[2],[5],[8]) are provably unused: feedback potential == 0.

  float* out     = (float*)d_out;               // 12288 floats
  float* trace   = (float*)d_ws;                // KTOT floats (reused per layer)
  float* partial = trace + KTOT;                // KTOT floats (KSPLIT x NN)

  float* p2 = out;                              // net_probas   [1,2048]
  float* s2 = out + NN;                         // net_outputs  [1,2048]
  float* p0 = out + 2 * NN;                     // probas_hidden[0]
  float* p1 = out + 3 * NN;                     // probas_hidden[1]
  float* s0 = out + 4 * NN;                     // outputs_hidden[0]
  float* s1 = out + 5 * NN;                     // outputs_hidden[1]

  snn_trace0<<<KTOT / 256, 256, 0, stream>>>(inp, trace);

  snn_gemv_wmma<<<NN / 16, 256, 0, stream>>>(ff_w0, trace, partial);
  snn_act<<<NN / 256, 256, 0, stream>>>(partial, b0, p0, s0, trace, 0xA001u);

  snn_gemv_wmma<<<NN / 16, 256, 0, stream>>>(ff_w1, trace, partial);
  snn_act<<<NN / 256, 256, 0, stream>>>(partial, b1, p1, s1, trace, 0xA002u);

  snn_gemv_wmma<<<NN / 16, 256, 0, stream>>>(ff_w2, trace, partial);
  snn_act<<<NN / 256, 256, 0, stream>>>(partial, b2, p2, s2, nullptr, 0xA003u);
}